// LearnedMRT_88819923681827
// MI455X (gfx1250) — compile-verified
//
#include <hip/hip_runtime.h>
#include <cstdint>

// CDNA5 (gfx1250) wave32 WMMA types
typedef __attribute__((ext_vector_type(2))) float v2f;  // f32 A/B operand: 16x4 / 4x16 = 2 VGPRs
typedef __attribute__((ext_vector_type(8))) float v8f;  // f32 C/D operand: 16x16 = 8 VGPRs

#define TAU_INV (1.0f / 0.7f)

// D = A(16x4) * B(4x16) + C, full fp32 precision on the matrix pipe.
__device__ __forceinline__ v8f wmma4(v2f a, v2f b, v8f c) {
    return __builtin_amdgcn_wmma_f32_16x16x4_f32(
        /*neg_a=*/false, a, /*neg_b=*/false, b,
        /*c_mod=*/(short)0, c, /*reuse_a=*/false, /*reuse_b=*/false);
}

static __device__ const int FLIP_d[9] = {0, 2, 1, 5, 4, 3, 6, 8, 7};

// W1 augmented with bias as K-column 9; rows >= 24 and K >= 10 are zero padding.
__device__ __forceinline__ float w1aug(const float* __restrict__ w1,
                                       const float* __restrict__ b1,
                                       bool flip, int row, int k) {
    if (row >= 24 || k >= 10) return 0.0f;
    if (k == 9) return b1[row];
    int kk = flip ? FLIP_d[k] : k;
    return w1[row * 9 + kk];
}

extern "C" __global__ void __launch_bounds__(256)
lmrt_wmma_kernel(const float* __restrict__ f,
                 const float* __restrict__ jw1, const float* __restrict__ jb1,
                 const float* __restrict__ jw2, const float* __restrict__ jb2,
                 const float* __restrict__ nw1, const float* __restrict__ nb1,
                 const float* __restrict__ nw2, const float* __restrict__ nb2,
                 const float* __restrict__ Mm,  const float* __restrict__ Mi,
                 float* __restrict__ out, long long Npix, int nGroups)
{
    const int lane = (int)(threadIdx.x & 31);
    const int pl   = lane & 15;     // pixel slot within group (N column)
    const int half = lane >> 4;     // 0: lanes 0-15, 1: lanes 16-31
    const int wid  = (int)((blockIdx.x * blockDim.x + threadIdx.x) >> 5);
    const int nWaves = (int)((gridDim.x * blockDim.x) >> 5);

    // ---- Hoisted A-layout weight tiles (loop-invariant, live in VGPRs) ----
    // f32 16x4 A layout: lane = row (M), VGPR q holds K = 4c + 2*half + q.
    v2f MA[3], MiA[3];
#pragma unroll
    for (int c = 0; c < 3; ++c) {
        int k0 = 4 * c + 2 * half;
        v2f a, b;
        a.x = (pl < 9 && k0     < 9) ? Mm[pl * 9 + k0]     : 0.0f;
        a.y = (pl < 9 && k0 + 1 < 9) ? Mm[pl * 9 + k0 + 1] : 0.0f;
        b.x = (pl < 9 && k0     < 9) ? Mi[pl * 9 + k0]     : 0.0f;
        b.y = (pl < 9 && k0 + 1 < 9) ? Mi[pl * 9 + k0 + 1] : 0.0f;
        MA[c] = a; MiA[c] = b;
    }

    // 4 W1 variants: {j, j_flip, n, n_flip}; 2 row-tiles (rows 0-15, 16-23+pad); 3 K-chunks.
    v2f W1A[4][2][3];
    const float* w1s[4] = {jw1, jw1, nw1, nw1};
    const float* b1s[4] = {jb1, jb1, nb1, nb1};
#pragma unroll
    for (int v = 0; v < 4; ++v) {
        bool flip = (v & 1) != 0;
#pragma unroll
        for (int t = 0; t < 2; ++t) {
            int row = pl + 16 * t;
#pragma unroll
            for (int c = 0; c < 3; ++c) {
                int k0 = 4 * c + 2 * half;
                v2f a;
                a.x = w1aug(w1s[v], b1s[v], flip, row, k0);
                a.y = w1aug(w1s[v], b1s[v], flip, row, k0 + 1);
                W1A[v][t][c] = a;
            }
        }
    }

    // Second-layer weights, lane-half selected (hidden unit u = k + 8*half / 16+k).
    float w2selJ[8], w2selN[8], w2hiJ[8], w2hiN[8];
#pragma unroll
    for (int k = 0; k < 8; ++k) {
        w2selJ[k] = jw2[k + 8 * half];
        w2selN[k] = nw2[k + 8 * half];
        w2hiJ[k]  = jw2[16 + k];
        w2hiN[k]  = nw2[16 + k];
    }
    const float jb2v = jb2[0], nb2v = nb2[0];

    // ---- Hoisted per-lane channel base pointers (branch-free B-operand loads) ----
    const size_t N = (size_t)Npix;
    const float* cq0a = f + (size_t)(2 * half) * N;       // K = 2h   (chunk 0, VGPR0)
    const float* cq1a = cq0a + N;                         // K = 2h+1 (chunk 0, VGPR1)
    const float* cq0b = cq0a + 4 * N;                     // K = 2h+4 (chunk 1, VGPR0)
    const float* cq1b = cq1a + 4 * N;                     // K = 2h+5 (chunk 1, VGPR1)
    const float* f8   = f + 8 * N;                        // channel 8 (uniform base)

    // Output base pointers: ob08 is per-lane (ch0 lanes lo / ch8 lanes hi); o[1..7] uniform.
    float* ob08 = out + (size_t)(8 * half) * N;
    float* oc1 = out + 1 * N;  float* oc2 = out + 2 * N;
    float* oc3 = out + 3 * N;  float* oc4 = out + 4 * N;
    float* oc5 = out + 5 * N;  float* oc6 = out + 6 * N;
    float* oc7 = out + 7 * N;

    for (int g = wid; g < nGroups; g += nWaves) {
        const int p = g * 16 + pl;
        // Clamped load index: only differs from p in a ragged final group (never for 2048^2).
        int pc = p;
        if ((long long)pc >= Npix) pc = (int)(Npix - 1);

        // ---- Stage 1: m(9x16) = M(9x9) * f(9x16); branch-free loads ----
        v8f macc = {0, 0, 0, 0, 0, 0, 0, 0};
        v2f b;
        b.x = cq0a[pc];  b.y = cq1a[pc];
        macc = wmma4(MA[0], b, macc);
        b.x = cq0b[pc];  b.y = cq1b[pc];
        macc = wmma4(MA[1], b, macc);
        const float f8v = f8[pc];                  // uniform base, full-wave load
        b.x = half ? 0.0f : f8v;  b.y = 0.0f;      // K=10,9,11 pads are zero
        macc = wmma4(MA[2], b, macc);

        // D layout: macc[v] lanes0-15 = m[v], lanes16-31 = m[v+8]. Broadcast to all lanes.
        float mf[9];
#pragma unroll
        for (int i = 0; i < 8; ++i) mf[i] = __shfl(macc[i], pl, 32);
        mf[8] = __shfl(macc[0], pl + 16, 32);

        // ---- Shared MLP B operand: m augmented with bias channel K=9 == 1.0 ----
        v2f hB[3];
        hB[0].x = half ? mf[2] : mf[0];  hB[0].y = half ? mf[3] : mf[1];
        hB[1].x = half ? mf[6] : mf[4];  hB[1].y = half ? mf[7] : mf[5];
        hB[2].x = half ? 0.0f  : mf[8];  hB[2].y = half ? 0.0f  : 1.0f;

        // ---- Hidden layers on WMMA, 24->1 reduction on VALU ----
        float dot[4];
#pragma unroll
        for (int v = 0; v < 4; ++v) {
            v8f h0 = {0, 0, 0, 0, 0, 0, 0, 0};   // hidden units 0-15
            v8f h1 = {0, 0, 0, 0, 0, 0, 0, 0};   // hidden units 16-23 (+zero pad)
#pragma unroll
            for (int c = 0; c < 3; ++c) {
                h0 = wmma4(W1A[v][0][c], hB[c], h0);
                h1 = wmma4(W1A[v][1][c], hB[c], h1);
            }
            float s = 0.0f;
#pragma unroll
            for (int k = 0; k < 8; ++k) {
                float w2a = (v < 2) ? w2selJ[k] : w2selN[k];
                float w2b = (v < 2) ? w2hiJ[k]  : w2hiN[k];
                s += fmaxf(h0[k], 0.0f) * w2a;   // lanes lo: u=k, lanes hi: u=8+k
                s += fmaxf(h1[k], 0.0f) * w2b;   // lanes lo: u=16+k, lanes hi: exact 0
            }
            s += __shfl_xor(s, 16, 32);          // combine lane-half partials
            dot[v] = s;
        }

        const float tjx = 0.5f + __expf(dot[0] + jb2v);
        const float tjy = 0.5f + __expf(dot[1] + jb2v);
        const float tn  = 0.5f + __expf(dot[2] + dot[3] + 2.0f * nb2v);

        // ---- Moment relaxation (per-lane scalar) ----
        const float rho = mf[0], jx = mf[1], jy = mf[2];
        const float ir = 1.0f / rho;
        float mp[9];
        mp[0] = rho; mp[1] = jx; mp[2] = jy;
        mp[3] = mf[3] - (mf[3] - jx * jx * ir) * TAU_INV;
        mp[4] = mf[4] - (mf[4] - jx * jy * ir) * TAU_INV;
        mp[5] = mf[5] - (mf[5] - jy * jy * ir) * TAU_INV;
        mp[6] = mf[6] * (1.0f - 1.0f / tn);
        mp[7] = mf[7] * (1.0f - 1.0f / tjx);
        mp[8] = mf[8] * (1.0f - 1.0f / tjy);

        // ---- Stage 3: out(9x16) = Minv(9x9) * m_post(9x16) ----
        v8f oacc = {0, 0, 0, 0, 0, 0, 0, 0};
        v2f oB;
        oB.x = half ? mp[2] : mp[0];  oB.y = half ? mp[3] : mp[1];
        oacc = wmma4(MiA[0], oB, oacc);
        oB.x = half ? mp[6] : mp[4];  oB.y = half ? mp[7] : mp[5];
        oacc = wmma4(MiA[1], oB, oacc);
        oB.x = half ? 0.0f  : mp[8];  oB.y = 0.0f;
        oacc = wmma4(MiA[2], oB, oacc);

        // ---- Stores: wave-uniform fast path; single exec toggle for ch1-7 ----
        const bool fullGroup = ((long long)g * 16 + 16 <= Npix);   // uniform per wave
        if (fullGroup) {
            ob08[p] = oacc[0];            // full wave: ch0 (lanes lo) + ch8 (lanes hi)
            if (half == 0) {              // one exec toggle for all seven stores
                oc1[p] = oacc[1];  oc2[p] = oacc[2];  oc3[p] = oacc[3];
                oc4[p] = oacc[4];  oc5[p] = oacc[5];  oc6[p] = oacc[6];
                oc7[p] = oacc[7];
            }
        } else {                          // ragged tail group (never taken for 2048^2)
            if ((long long)p < Npix) {
                ob08[p] = oacc[0];
                if (half == 0) {
                    oc1[p] = oacc[1];  oc2[p] = oacc[2];  oc3[p] = oacc[3];
                    oc4[p] = oacc[4];  oc5[p] = oacc[5];  oc6[p] = oacc[6];
                    oc7[p] = oacc[7];
                }
            }
        }
    }
}

extern "C" void kernel_launch(void* const* d_in, const int* in_sizes, int n_in,
                              void* d_out, int out_size, void* d_ws, size_t ws_size,
                              hipStream_t stream) {
    const float* f   = (const float*)d_in[0];
    const float* jw1 = (const float*)d_in[1];
    const float* jb1 = (const float*)d_in[2];
    const float* jw2 = (const float*)d_in[3];
    const float* jb2 = (const float*)d_in[4];
    const float* nw1 = (const float*)d_in[5];
    const float* nb1 = (const float*)d_in[6];
    const float* nw2 = (const float*)d_in[7];
    const float* nb2 = (const float*)d_in[8];
    const float* Mm  = (const float*)d_in[9];
    const float* Mi  = (const float*)d_in[10];
    float* out = (float*)d_out;

    const long long Npix = (long long)in_sizes[0] / 9;      // 2048*2048
    const int nGroups = (int)((Npix + 15) / 16);

    // 8 waves per 256-thread block; target ~8 groups per wave (amortize weight setup).
    int wavesTarget = (nGroups + 7) / 8;
    int blocks = (wavesTarget + 7) / 8;
    if (blocks < 1) blocks = 1;
    if (blocks > 8192) blocks = 8192;

    lmrt_wmma_kernel<<<blocks, 256, 0, stream>>>(
        f, jw1, jb1, jw2, jb2, nw1, nb1, nw2, nb2, Mm, Mi, out, Npix, nGroups);
}